// Global_update_Layer_32822140076360
// MI455X (gfx1250) — compile-verified
//
#include <hip/hip_runtime.h>
#include <math.h>

typedef __attribute__((ext_vector_type(2))) float v2f;
typedef __attribute__((ext_vector_type(8))) float v8f;

#define CB      16          // C*B
#define NN      1024        // N
#define EE      16384       // E
#define DMODEL  64          // DIM
#define NH      4           // heads
#define DHD     16          // DIM / H

static __device__ __forceinline__ v8f wmma_f32(v2f a, v2f b, v8f c) {
  // V_WMMA_F32_16X16X4_F32 : D = A(16x4) * B(4x16) + C(16x16), all f32, wave32
  return __builtin_amdgcn_wmma_f32_16x16x4_f32(false, a, false, b, (short)0, c, false, false);
}
static __device__ __forceinline__ float silu_f(float x) { return x / (1.0f + __expf(-x)); }

// CDNA5 async Global->LDS copy (ASYNCcnt-tracked), see CDNA5 ISA
static __device__ __forceinline__ void async_load_b128(unsigned int lds_byte_off,
                                                       const float* gaddr) {
  asm volatile("global_load_async_to_lds_b128 %0, %1, off"
               :: "v"(lds_byte_off), "v"(gaddr) : "memory");
}
static __device__ __forceinline__ void wait_async0() {
  asm volatile("s_wait_asynccnt 0" ::: "memory");
}

// ---------------------------------------------------------------------------
// Kernel 1: six projections  out[m] = xh @ W_m + b_m   (W: [64,64] row-major)
// one wave per (matrix, 16-row tile, 16-col tile); 6*1024*4 = 24576 waves
// ---------------------------------------------------------------------------
struct ProjArgs {
  const float* xh;
  const float* w[6];
  const float* b[6];
  float*       o[6];
};

__global__ void proj_kernel(ProjArgs pa) {
  int wid  = (blockIdx.x * blockDim.x + threadIdx.x) >> 5;
  int lane = threadIdx.x & 31;
  int mat  = wid / 4096;
  int rem  = wid % 4096;
  int mt   = rem >> 2;
  int nt   = rem & 3;
  int m    = lane & 15, hi = lane >> 4;

  const float* arow = pa.xh + (size_t)(mt * 16 + m) * DMODEL;
  const float* W    = pa.w[mat];
  v8f acc = {};
#pragma unroll
  for (int k0 = 0; k0 < DMODEL; k0 += 4) {
    int ka = k0 + 2 * hi;
    v2f a;  a.x  = arow[ka];            a.y  = arow[ka + 1];
    const float* wp = W + (size_t)ka * DMODEL + nt * 16 + m;
    v2f bb; bb.x = wp[0];               bb.y = wp[DMODEL];
    acc = wmma_f32(a, bb, acc);
  }
  float bias = pa.b[mat][nt * 16 + m];
  float* out = pa.o[mat];
#pragma unroll
  for (int r = 0; r < 8; ++r) {
    int row = mt * 16 + r + 8 * hi;
    out[(size_t)row * DMODEL + nt * 16 + m] = acc[r] + bias;
  }
}

// ---------------------------------------------------------------------------
// Kernel 2: per-edge message + scatter-add into res_v [CB, N, 3, 64]
// ---------------------------------------------------------------------------
__global__ void edge_kernel(const float* __restrict__ q1, const float* __restrict__ k1,
                            const float* __restrict__ v1, const float* __restrict__ x,
                            const int* __restrict__ ei, const int* __restrict__ ej,
                            float* __restrict__ res_v) {
  int t = blockIdx.x * blockDim.x + threadIdx.x;
  if (t >= CB * EE) return;
  int cb = t / EE, e = t % EE;
  int i = ei[e], j = ej[e];
  const float* qr = q1 + (size_t)(cb * NN + i) * DMODEL;
  const float* kr = k1 + (size_t)(cb * NN + j) * DMODEL;
  const float* vr = v1 + (size_t)(cb * NN + j) * DMODEL;
  const float invd = 1.0f / (4.0f + 1e-7f);
  float w[NH];
#pragma unroll
  for (int h = 0; h < NH; ++h) {
    float s = 0.f;
#pragma unroll
    for (int f = 0; f < DHD; ++f) s += qr[h * DHD + f] * kr[h * DHD + f];
    w[h] = tanhf(s * invd);
  }
  const float* xi = x + (size_t)(cb * NN + i) * 3;
  const float* xj = x + (size_t)(cb * NN + j) * 3;
  float dx0 = xj[0] - xi[0], dx1 = xj[1] - xi[1], dx2 = xj[2] - xi[2];
  float* base = res_v + (size_t)(cb * NN + i) * 3 * DMODEL;
#pragma unroll 4
  for (int c2 = 0; c2 < DMODEL; ++c2) {
    float wv = w[c2 >> 4] * vr[c2];
    atomicAdd(base + 0 * DMODEL + c2, wv * dx0);
    atomicAdd(base + 1 * DMODEL + c2, wv * dx1);
    atomicAdd(base + 2 * DMODEL + c2, wv * dx2);
  }
}

// ---------------------------------------------------------------------------
// Kernel 3: g_delta_v = (SiLU(res_v @ w1 + b1)) @ w2 + b2   rows = 49152
// ---------------------------------------------------------------------------
__global__ void netv_kernel(const float* __restrict__ res_v,
                            const float* __restrict__ w1, const float* __restrict__ b1,
                            const float* __restrict__ w2, const float* __restrict__ b2,
                            float* __restrict__ gdv) {
  int wid  = (blockIdx.x * blockDim.x + threadIdx.x) >> 5;   // tile 0..3071
  int lane = threadIdx.x & 31;
  int m = lane & 15, hi = lane >> 4;
  const float* arow = res_v + (size_t)(wid * 16 + m) * DMODEL;
  float part[8] = {0.f, 0.f, 0.f, 0.f, 0.f, 0.f, 0.f, 0.f};
#pragma unroll
  for (int nt = 0; nt < 4; ++nt) {
    v8f acc = {};
#pragma unroll
    for (int k0 = 0; k0 < DMODEL; k0 += 4) {
      int ka = k0 + 2 * hi;
      v2f a;  a.x  = arow[ka];  a.y = arow[ka + 1];
      const float* wp = w1 + (size_t)ka * DMODEL + nt * 16 + m;
      v2f bb; bb.x = wp[0];     bb.y = wp[DMODEL];
      acc = wmma_f32(a, bb, acc);
    }
    float bn  = b1[nt * 16 + m];
    float w2n = w2[nt * 16 + m];
#pragma unroll
    for (int r = 0; r < 8; ++r) part[r] += silu_f(acc[r] + bn) * w2n;
  }
#pragma unroll
  for (int r = 0; r < 8; ++r) {
    part[r] += __shfl_xor(part[r], 1, 32);
    part[r] += __shfl_xor(part[r], 2, 32);
    part[r] += __shfl_xor(part[r], 4, 32);
    part[r] += __shfl_xor(part[r], 8, 32);
  }
  if (m == 0) {
    float bb2 = b2[0];
#pragma unroll
    for (int r = 0; r < 8; ++r) gdv[wid * 16 + r + 8 * hi] = part[r] + bb2;
  }
}

// ---------------------------------------------------------------------------
// Kernel 4: dense attention  res_x = (tanh(Q K^T / d) * d) @ V  per (cb,h)
// block = (cb, h, 8 q-blocks); K/V tiles double-buffered in LDS via
// GLOBAL_LOAD_ASYNC_TO_LDS_B128 (shared by all 8 waves); per-wave D->A via LDS
// ---------------------------------------------------------------------------
__global__ void attn_kernel(const float* __restrict__ q2, const float* __restrict__ k2,
                            const float* __restrict__ v2, float* __restrict__ res_x) {
  // layout (floats): kbuf[2][16*20] @0, vbuf[2][16*20] @640, sbuf[8][16*17] @1280
  __shared__ float smem[2 * 320 + 2 * 320 + 8 * 272];
  const int VB0 = 640, SB0 = 1280;
  int tid   = threadIdx.x;
  int wslot = tid >> 5;
  int lane  = tid & 31;
  int bid = blockIdx.x;            // 512 blocks
  int cb  = bid >> 5;
  int rem = bid & 31;
  int h   = rem >> 3;
  int qg  = rem & 7;
  int qb  = qg * 8 + wslot;
  int m = lane & 15, hi = lane >> 4;
  const float dsc  = 4.0f + 1e-7f;
  const float invd = 1.0f / dsc;

  // async stage of K/V tile jb into buffer buf: 16 rows x 64B, one b128/lane
  auto issue = [&](int jb, int buf) {
    if (tid < 64) {                       // waves 0,1 -> K tile
      int row = tid >> 2, seg = tid & 3;
      const float* g = k2 + (size_t)(cb * NN + jb * 16 + row) * DMODEL + h * DHD + seg * 4;
      unsigned int off = (unsigned int)(buf * 320 + row * 20 + seg * 4) * 4u;
      async_load_b128(off, g);
    } else if (tid < 128) {               // waves 2,3 -> V tile
      int t2 = tid - 64;
      int row = t2 >> 2, seg = t2 & 3;
      const float* g = v2 + (size_t)(cb * NN + jb * 16 + row) * DMODEL + h * DHD + seg * 4;
      unsigned int off = (unsigned int)(VB0 + buf * 320 + row * 20 + seg * 4) * 4u;
      async_load_b128(off, g);
    }
  };

  const float* Qrow = q2 + (size_t)(cb * NN + qb * 16 + m) * DMODEL + h * DHD;
  float qa[8];
#pragma unroll
  for (int s = 0; s < 4; ++s) {
    int ka = 4 * s + 2 * hi;
    qa[2 * s] = Qrow[ka];  qa[2 * s + 1] = Qrow[ka + 1];
  }
  float* sl = smem + SB0 + wslot * 272;
  v8f oacc = {};
  issue(0, 0);
  for (int jb = 0; jb < 64; ++jb) {
    int cur = jb & 1;
    wait_async0();                 // own async copies for buf[cur] landed
    __syncthreads();               // whole block sees buf[cur]; prev reads of buf[1-cur] done
    if (jb < 63) issue(jb + 1, 1 - cur);   // overlap next tile copy with compute
    const float* kb = smem + cur * 320;
    const float* vb = smem + VB0 + cur * 320;
    v8f sacc = {};
#pragma unroll
    for (int s = 0; s < 4; ++s) {
      int ka = 4 * s + 2 * hi;
      v2f a;  a.x  = qa[2 * s];         a.y  = qa[2 * s + 1];
      v2f bb; bb.x = kb[m * 20 + ka];   bb.y = kb[m * 20 + ka + 1];
      sacc = wmma_f32(a, bb, sacc);
    }
#pragma unroll
    for (int r = 0; r < 8; ++r)
      sl[(r + 8 * hi) * 17 + m] = tanhf(sacc[r] * invd) * dsc;   // D-layout store
    // LDS ops from one wave complete in order -> safe D->A relayout
#pragma unroll
    for (int s = 0; s < 4; ++s) {
      int ka = 4 * s + 2 * hi;
      v2f a;  a.x  = sl[m * 17 + ka];      a.y  = sl[m * 17 + ka + 1];
      v2f bb; bb.x = vb[ka * 20 + m];      bb.y = vb[(ka + 1) * 20 + m];
      oacc = wmma_f32(a, bb, oacc);
    }
  }
#pragma unroll
  for (int r = 0; r < 8; ++r)
    res_x[(size_t)(cb * NN + qb * 16 + r + 8 * hi) * DMODEL + h * DHD + m] = oacc[r];
}

// ---------------------------------------------------------------------------
// Kernel 5: cat = [xh, res_x];  g_h = net2h(cat);  g_delta_x = (v+gdv)*net2x(cat)
// ---------------------------------------------------------------------------
__global__ void final_kernel(const float* __restrict__ xh, const float* __restrict__ res_x,
                             const float* __restrict__ vin, const float* __restrict__ gdv,
                             const float* __restrict__ h_w1, const float* __restrict__ h_b1,
                             const float* __restrict__ h_w2, const float* __restrict__ h_b2,
                             const float* __restrict__ x_w1, const float* __restrict__ x_b1,
                             const float* __restrict__ x_w2, const float* __restrict__ x_b2,
                             float* __restrict__ out_dx, float* __restrict__ out_h) {
  __shared__ float smem[8 * 16 * 68];
  int wslot = threadIdx.x >> 5;
  int lane  = threadIdx.x & 31;
  int t     = (blockIdx.x * blockDim.x + threadIdx.x) >> 5;   // tile 0..1023
  int m = lane & 15, hi = lane >> 4;
  int row0 = t * 16;
  const float* xrow = xh    + (size_t)(row0 + m) * DMODEL;
  const float* rrow = res_x + (size_t)(row0 + m) * DMODEL;
  float* hh = smem + wslot * (16 * 68);
  float part[8] = {0.f, 0.f, 0.f, 0.f, 0.f, 0.f, 0.f, 0.f};

#pragma unroll
  for (int nt = 0; nt < 4; ++nt) {
    v8f acch = {};
    v8f accx = {};
#pragma unroll
    for (int k0 = 0; k0 < 2 * DMODEL; k0 += 4) {
      int ka = k0 + 2 * hi;
      v2f a;
      if (ka < DMODEL) { a.x = xrow[ka];          a.y = xrow[ka + 1]; }
      else             { a.x = rrow[ka - DMODEL]; a.y = rrow[ka - DMODEL + 1]; }
      const float* wph = h_w1 + (size_t)ka * DMODEL + nt * 16 + m;
      v2f bh; bh.x = wph[0];  bh.y = wph[DMODEL];
      acch = wmma_f32(a, bh, acch);
      const float* wpx = x_w1 + (size_t)ka * DMODEL + nt * 16 + m;
      v2f bx; bx.x = wpx[0];  bx.y = wpx[DMODEL];
      accx = wmma_f32(a, bx, accx);
    }
    float bh1 = h_b1[nt * 16 + m];
    float bx1 = x_b1[nt * 16 + m];
    float xw2 = x_w2[nt * 16 + m];
#pragma unroll
    for (int r = 0; r < 8; ++r) {
      hh[(r + 8 * hi) * 68 + nt * 16 + m] = silu_f(acch[r] + bh1);
      part[r] += silu_f(accx[r] + bx1) * xw2;
    }
  }
  // second layer of net2h: g_h = hh @ h_w2 + h_b2  (hh re-read in A-layout)
#pragma unroll
  for (int nt = 0; nt < 4; ++nt) {
    v8f acc = {};
#pragma unroll
    for (int k0 = 0; k0 < DMODEL; k0 += 4) {
      int ka = k0 + 2 * hi;
      v2f a;  a.x  = hh[m * 68 + ka];  a.y = hh[m * 68 + ka + 1];
      const float* wp = h_w2 + (size_t)ka * DMODEL + nt * 16 + m;
      v2f bb; bb.x = wp[0];  bb.y = wp[DMODEL];
      acc = wmma_f32(a, bb, acc);
    }
    float b2 = h_b2[nt * 16 + m];
#pragma unroll
    for (int r = 0; r < 8; ++r)
      out_h[(size_t)(row0 + r + 8 * hi) * DMODEL + nt * 16 + m] = acc[r] + b2;
  }
  // net2x scalar + g_delta_x
#pragma unroll
  for (int r = 0; r < 8; ++r) {
    part[r] += __shfl_xor(part[r], 1, 32);
    part[r] += __shfl_xor(part[r], 2, 32);
    part[r] += __shfl_xor(part[r], 4, 32);
    part[r] += __shfl_xor(part[r], 8, 32);
  }
  float b2x = x_b2[0];
  if (m < 3) {
#pragma unroll
    for (int r = 0; r < 8; ++r) {
      int row = row0 + r + 8 * hi;
      out_dx[(size_t)row * 3 + m] =
          (vin[(size_t)row * 3 + m] + gdv[(size_t)row * 3 + m]) * (part[r] + b2x);
    }
  }
}

// ---------------------------------------------------------------------------
extern "C" void kernel_launch(void* const* d_in, const int* in_sizes, int n_in,
                              void* d_out, int out_size, void* d_ws, size_t ws_size,
                              hipStream_t stream) {
  (void)in_sizes; (void)n_in; (void)out_size; (void)ws_size;
  const float* x    = (const float*)d_in[0];
  const float* xh   = (const float*)d_in[1];
  const int*   e    = (const int*)d_in[2];
  const float* vin  = (const float*)d_in[4];
  const float* pw[6], *pb[6];
  for (int i = 0; i < 6; ++i) { pw[i] = (const float*)d_in[5 + 2 * i]; pb[i] = (const float*)d_in[6 + 2 * i]; }
  const float* netv_w1 = (const float*)d_in[17];
  const float* netv_b1 = (const float*)d_in[18];
  const float* netv_w2 = (const float*)d_in[19];
  const float* netv_b2 = (const float*)d_in[20];
  const float* h_w1 = (const float*)d_in[21];
  const float* h_b1 = (const float*)d_in[22];
  const float* h_w2 = (const float*)d_in[23];
  const float* h_b2 = (const float*)d_in[24];
  const float* x_w1 = (const float*)d_in[25];
  const float* x_b1 = (const float*)d_in[26];
  const float* x_w2 = (const float*)d_in[27];
  const float* x_b2 = (const float*)d_in[28];

  const size_t MAT = (size_t)CB * NN * DMODEL;          // 1,048,576 floats
  float* ws    = (float*)d_ws;
  float* proj[6];
  for (int i = 0; i < 6; ++i) proj[i] = ws + i * MAT;   // q1,k1,v1,q2,k2,v2
  float* res_v = ws + 6 * MAT;                          // 3,145,728 floats
  float* gdv   = res_v + (size_t)CB * NN * 3 * DMODEL;  //    49,152 floats
  float* res_x = gdv + (size_t)CB * NN * 3;             // 1,048,576 floats

  ProjArgs pa;
  pa.xh = xh;
  for (int i = 0; i < 6; ++i) { pa.w[i] = pw[i]; pa.b[i] = pb[i]; pa.o[i] = proj[i]; }
  proj_kernel<<<3072, 256, 0, stream>>>(pa);

  hipMemsetAsync(res_v, 0, (size_t)CB * NN * 3 * DMODEL * sizeof(float), stream);
  edge_kernel<<<1024, 256, 0, stream>>>(proj[0], proj[1], proj[2], x, e, e + EE, res_v);
  netv_kernel<<<384, 256, 0, stream>>>(res_v, netv_w1, netv_b1, netv_w2, netv_b2, gdv);
  attn_kernel<<<512, 256, 0, stream>>>(proj[3], proj[4], proj[5], res_x);

  float* out_dx = (float*)d_out;
  float* out_h  = out_dx + (size_t)CB * NN * 3;
  final_kernel<<<128, 256, 0, stream>>>(xh, res_x, vin, gdv,
                                        h_w1, h_b1, h_w2, h_b2,
                                        x_w1, x_b1, x_w2, x_b2,
                                        out_dx, out_h);
}